// VolumeNormalizer_61753039782329
// MI455X (gfx1250) — compile-verified
//
#include <hip/hip_runtime.h>
#include <hip/hip_bf16.h>

// ---------------- problem constants (from the reference) ----------------
#define BB   32
#define VV   20000
#define TT   40000
#define NN1  1500
#define NN2  1000
#define QP_ITERS 60

// ---------------- GEMM tiling ----------------
#define KC      64     // K chunk staged through LDS
#define LDA     68     // padded LDS row stride (floats): lane stride 68 -> banks spread
#define GWAVES  8      // waves per block (256 threads)
#define KSPLIT  25     // grid split of K; TT/KSPLIT = 1600 = 25*KC

typedef float v2f __attribute__((ext_vector_type(2)));
typedef float v8f __attribute__((ext_vector_type(8)));

// ---------------------------------------------------------------------------
// pz construction
// ---------------------------------------------------------------------------
__global__ void k_fill_pz(const float* __restrict__ tz, float* __restrict__ pz, int n) {
  int i = blockIdx.x * blockDim.x + threadIdx.x;
  if (i < n) pz[i] = tz[i % (VV * 3)];
}

__global__ void k_scatter_y(const float* __restrict__ y, const int* __restrict__ idx2,
                            float* __restrict__ pz) {
  int i = blockIdx.x * blockDim.x + threadIdx.x;
  if (i >= BB * NN2) return;
  int b = i / NN2, n = i - b * NN2;
  int v = idx2[n];
  float* p = pz + ((size_t)b * VV + v) * 3;
  p[0] = y[b * (2 * NN2) + 2 * n + 0];
  p[2] = y[b * (2 * NN2) + 2 * n + 1];
}

__global__ void k_scatter_x(const float* __restrict__ x, const int* __restrict__ idx1,
                            float* __restrict__ pz) {
  int i = blockIdx.x * blockDim.x + threadIdx.x;
  if (i >= BB * NN1 * 3) return;
  int c = i % 3;
  int n = (i / 3) % NN1;
  int b = i / (3 * NN1);
  int v = idx1[n];
  pz[((size_t)b * VV + v) * 3 + c] = x[i];
}

__global__ void k_zero(float* __restrict__ p, int n) {
  int i = blockIdx.x * blockDim.x + threadIdx.x;
  if (i < n) p[i] = 0.0f;
}

// ---------------------------------------------------------------------------
// volume reduction:  vol[bb] += sum_t (x0+x1+x2) * det/6 ;  bb==BB -> temp_zero
// ---------------------------------------------------------------------------
__global__ __launch_bounds__(256) void k_volume(const float* __restrict__ pz,
                                                const float* __restrict__ tz,
                                                const int* __restrict__ tri,
                                                float* __restrict__ vol) {
  __shared__ float red[256];
  int t  = blockIdx.x * 256 + threadIdx.x;
  int bb = blockIdx.y;
  float contrib = 0.0f;
  if (t < TT) {
    int i0 = tri[t * 3 + 0], i1 = tri[t * 3 + 1], i2 = tri[t * 3 + 2];
    const float* base = (bb == BB) ? tz : pz + (size_t)bb * VV * 3;
    const float* p0 = base + (size_t)i0 * 3;
    const float* p1 = base + (size_t)i1 * 3;
    const float* p2 = base + (size_t)i2 * 3;
    float d = ((p1[1] - p0[1]) * (p2[2] - p0[2]) -
               (p1[2] - p0[2]) * (p2[1] - p0[1])) * (1.0f / 6.0f);
    contrib = (p0[0] + p1[0] + p2[0]) * d;
  }
  red[threadIdx.x] = contrib;
  __syncthreads();
  for (int s = 128; s > 0; s >>= 1) {
    if (threadIdx.x < s) red[threadIdx.x] += red[threadIdx.x + s];
    __syncthreads();
  }
  if (threadIdx.x == 0) atomicAdd(&vol[bb], red[0]);
}

__global__ void k_compute_a(const float* __restrict__ vol, float* __restrict__ a) {
  int i = threadIdx.x;
  if (i < BB) a[i] = (vol[BB] - vol[i]) * (1.0f / 3.0f);
}

// ---------------------------------------------------------------------------
// per-triangle determinant for one axis (A-matrix of the GEMM), shape (B, T)
//   axis 2 (coeff_z): det([[x0-x2, y0-y2],[x1-x2, y1-y2]])/6
//   axis 1 (coeff_y): det([[x0-x1, z0-z1],[x2-x1, z2-z1]])/6
//   axis 0 (coeff_x): det([[y1-y0, z1-z0],[y2-y0, z2-z0]])/6
// ---------------------------------------------------------------------------
__global__ void k_det(const float* __restrict__ pz, const int* __restrict__ tri,
                      float* __restrict__ d, int axis) {
  const int i = blockIdx.x * blockDim.x + threadIdx.x;
  if (i >= BB * TT) return;
  const int b = i / TT;
  const int t = i - b * TT;
  const int i0 = tri[t * 3 + 0], i1 = tri[t * 3 + 1], i2 = tri[t * 3 + 2];
  const float* base = pz + (size_t)b * VV * 3;
  const float* p0 = base + (size_t)i0 * 3;
  const float* p1 = base + (size_t)i1 * 3;
  const float* p2 = base + (size_t)i2 * 3;
  float det;
  if (axis == 2) {
    det = (p0[0] - p2[0]) * (p1[1] - p2[1]) - (p0[1] - p2[1]) * (p1[0] - p2[0]);
  } else if (axis == 1) {
    det = (p0[0] - p1[0]) * (p2[2] - p1[2]) - (p0[2] - p1[2]) * (p2[0] - p1[0]);
  } else {
    det = (p1[1] - p0[1]) * (p2[2] - p0[2]) - (p1[2] - p0[2]) * (p2[1] - p0[1]);
  }
  d[i] = det * (1.0f / 6.0f);
}

// ---------------------------------------------------------------------------
// WMMA GEMM:  C(32 x NN1) += A(32 x TT) * Vf^T(TT x NN1)
//   One wave owns one 16-column N tile; M=32 -> two 16x16 f32 accumulators.
//   K grid-split KSPLIT ways; partials merged with global_atomic_add_f32.
//   V_WMMA_F32_16X16X4_F32 fragment layouts per CDNA5 ISA 7.12.2:
//     A 16x4:  lanes 0-15 hold M=lane, (v0,v1)=(K0,K1); lanes 16-31 -> (K2,K3)
//     B 4x16:  lanes 0-15 hold N=lane, (v0,v1)=(K0,K1); lanes 16-31 -> (K2,K3)
//     C 16x16: VGPR r: lanes 0-15 M=r, lanes 16-31 M=r+8, N=lane&15
// ---------------------------------------------------------------------------
__global__ __launch_bounds__(256) void k_gemm_wmma(const float* __restrict__ A,
                                                   const float* __restrict__ Vf,
                                                   float* __restrict__ C) {
  __shared__ float sA[32 * LDA];
  __shared__ float sB[GWAVES][16 * LDA];

  const int tid   = threadIdx.x;
  const int wave  = tid >> 5;
  const int lane  = tid & 31;
  const int lhalf = lane & 15;
  const int khi   = (lane >> 4) << 1;          // 0 (lanes 0-15) or 2 (lanes 16-31)
  const int n0    = (blockIdx.x * GWAVES + wave) * 16;
  const int kbeg  = blockIdx.y * (TT / KSPLIT);
  const int kend  = kbeg + (TT / KSPLIT);

  v8f acc0 = {};
  v8f acc1 = {};

  for (int k0 = kbeg; k0 < kend; k0 += KC) {
    // Stage A chunk (32 x KC): 512 float4 across 256 threads, coalesced.
    for (int j = tid; j < 512; j += 256) {
      const int row = j >> 4;
      const int col = (j & 15) << 2;
      const float4 v = *reinterpret_cast<const float4*>(A + (size_t)row * TT + k0 + col);
      *reinterpret_cast<float4*>(&sA[row * LDA + col]) = v;
    }
    // Stage this wave's 16 x KC tile of Vf (rows = output columns n0..n0+15).
    for (int j = lane; j < 256; j += 32) {
      const int row = j >> 4;
      const int col = (j & 15) << 2;
      const int n   = n0 + row;
      float4 v = {0.0f, 0.0f, 0.0f, 0.0f};
      if (n < NN1) {
        const float* src = Vf + (size_t)n * TT + k0 + col;
        v = *reinterpret_cast<const float4*>(src);
        if (k0 + KC < TT) __builtin_prefetch(src + KC, 0, 0);  // global_prefetch_b8
      }
      *reinterpret_cast<float4*>(&sB[wave][row * LDA + col]) = v;
    }
    __syncthreads();

#pragma unroll
    for (int ks = 0; ks < KC; ks += 4) {
      v2f bf, a0, a1;
      bf.x = sB[wave][lhalf * LDA + ks + khi + 0];
      bf.y = sB[wave][lhalf * LDA + ks + khi + 1];
      a0.x = sA[lhalf * LDA + ks + khi + 0];
      a0.y = sA[lhalf * LDA + ks + khi + 1];
      a1.x = sA[(16 + lhalf) * LDA + ks + khi + 0];
      a1.y = sA[(16 + lhalf) * LDA + ks + khi + 1];
      acc0 = __builtin_amdgcn_wmma_f32_16x16x4_f32(false, a0, false, bf, (short)0,
                                                   acc0, false, false);
      acc1 = __builtin_amdgcn_wmma_f32_16x16x4_f32(false, a1, false, bf, (short)0,
                                                   acc1, false, false);
    }
    __syncthreads();
  }

  const int n = n0 + lhalf;
  if (n < NN1) {
    const int mb = (lane >> 4) << 3;  // +8 rows for high half-wave
#pragma unroll
    for (int r = 0; r < 8; ++r) {
      atomicAdd(&C[(size_t)(r + mb) * NN1 + n], acc0[r]);
      atomicAdd(&C[(size_t)(16 + r + mb) * NN1 + n], acc1[r]);
    }
  }
}

// ---------------------------------------------------------------------------
// QP solve (one block per batch): 60 Newton iterations on scalar nu.
//   c = C[b, :], h = x[b, :, axis], a = a[b]; def = max(-nu*c, -h)
// ---------------------------------------------------------------------------
__global__ __launch_bounds__(256) void k_solve(const float* __restrict__ C,
                                               const float* __restrict__ x,
                                               const float* __restrict__ a,
                                               float* __restrict__ def, int axis) {
  __shared__ float2 red[256];
  const int b   = blockIdx.x;
  const int tid = threadIdx.x;

  float c[6], h[6];
#pragma unroll
  for (int j = 0; j < 6; ++j) {
    int n = tid + j * 256;
    if (n < NN1) {
      c[j] = C[(size_t)b * NN1 + n];
      h[j] = x[((size_t)b * NN1 + n) * 3 + axis];
    } else {
      c[j] = 0.0f;
      h[j] = 0.0f;
    }
  }
  const float ab = a[b];
  float nu = 0.0f;

  for (int it = 0; it < QP_ITERS; ++it) {
    float gs = 0.0f, gps = 0.0f;
#pragma unroll
    for (int j = 0; j < 6; ++j) {
      float znu = -nu * c[j];
      float z = fmaxf(znu, -h[j]);
      gs += c[j] * z;
      if (znu >= -h[j]) gps += c[j] * c[j];
    }
    red[tid] = make_float2(gs, gps);
    __syncthreads();
    for (int s = 128; s > 0; s >>= 1) {
      if (tid < s) {
        red[tid].x += red[tid + s].x;
        red[tid].y += red[tid + s].y;
      }
      __syncthreads();
    }
    float g  = red[0].x - ab;
    float gp = fminf(-red[0].y, -1e-12f);
    nu = nu - g / gp;
    __syncthreads();
  }

#pragma unroll
  for (int j = 0; j < 6; ++j) {
    int n = tid + j * 256;
    if (n < NN1) def[(size_t)b * NN1 + n] = fmaxf(-nu * c[j], -h[j]);
  }
}

// ---------------------------------------------------------------------------
// pz updates between solves
// ---------------------------------------------------------------------------
__global__ void k_update_add_z(const int* __restrict__ idx1, const float* __restrict__ dz,
                               float* __restrict__ pz) {
  int i = blockIdx.x * blockDim.x + threadIdx.x;
  if (i >= BB * NN1) return;
  int b = i / NN1, n = i - b * NN1;
  int v = idx1[n];
  pz[((size_t)b * VV + v) * 3 + 2] += dz[i];
}

// Reference quirk: pz[...,2] = pz[...,1] + def_y  (z coord set from y coord)
__global__ void k_update_set_zy(const int* __restrict__ idx1, const float* __restrict__ dy,
                                float* __restrict__ pz) {
  int i = blockIdx.x * blockDim.x + threadIdx.x;
  if (i >= BB * NN1) return;
  int b = i / NN1, n = i - b * NN1;
  int v = idx1[n];
  float* p = pz + ((size_t)b * VV + v) * 3;
  p[2] = p[1] + dy[i];
}

__global__ void k_output(const float* __restrict__ x, const float* __restrict__ dx,
                         const float* __restrict__ dy, const float* __restrict__ dz,
                         float* __restrict__ out) {
  int i = blockIdx.x * blockDim.x + threadIdx.x;
  if (i >= BB * NN1) return;
  out[i * 3 + 0] = x[i * 3 + 0] + dx[i];
  out[i * 3 + 1] = x[i * 3 + 1] + dy[i];
  out[i * 3 + 2] = x[i * 3 + 2] + dz[i];
}

// ---------------------------------------------------------------------------
extern "C" void kernel_launch(void* const* d_in, const int* in_sizes, int n_in,
                              void* d_out, int out_size, void* d_ws, size_t ws_size,
                              hipStream_t stream) {
  const float* x   = (const float*)d_in[0];  // (B, N1, 3)
  const float* y   = (const float*)d_in[1];  // (B, 2*N2)
  const float* tz  = (const float*)d_in[2];  // (V, 3)
  const float* vf  = (const float*)d_in[3];  // (N1, T)
  const int* tri   = (const int*)d_in[4];    // (T, 3)
  const int* idx1  = (const int*)d_in[5];    // (N1)
  const int* idx2  = (const int*)d_in[6];    // (N2)
  float* out = (float*)d_out;

  // workspace layout (floats), ~13.6 MB total, all 16B-aligned offsets
  float* ws   = (float*)d_ws;
  float* pz   = ws;                                   // B*V*3
  float* dmat = pz + (size_t)BB * VV * 3;             // B*T
  float* Cc   = dmat + (size_t)BB * TT;               // B*N1
  float* vol  = Cc + (size_t)BB * NN1;                // B+1 (padded 48)
  float* aarr = vol + 48;                             // B (padded 32)
  float* defx = aarr + 32;                            // B*N1
  float* defy = defx + (size_t)BB * NN1;              // B*N1
  float* defz = defy + (size_t)BB * NN1;              // B*N1

  auto cdiv = [](int a, int b) { return (a + b - 1) / b; };
  const int BT  = BB * TT;
  const int BN  = BB * NN1;
  const dim3 ggrid(cdiv(NN1, 16 * GWAVES) + ((NN1 % (16 * GWAVES)) ? 0 : 0), KSPLIT);
  const dim3 gemm_grid(12, KSPLIT);  // 12*8 waves cover 96 N-tiles >= ceil(1500/16)=94

  // build pz
  k_fill_pz<<<cdiv(BB * VV * 3, 256), 256, 0, stream>>>(tz, pz, BB * VV * 3);
  k_scatter_y<<<cdiv(BB * NN2, 256), 256, 0, stream>>>(y, idx2, pz);
  k_scatter_x<<<cdiv(BB * NN1 * 3, 256), 256, 0, stream>>>(x, idx1, pz);

  // volumes and a
  k_zero<<<1, 64, 0, stream>>>(vol, 48);
  k_volume<<<dim3(cdiv(TT, 256), BB + 1), 256, 0, stream>>>(pz, tz, tri, vol);
  k_compute_a<<<1, 64, 0, stream>>>(vol, aarr);

  // ---- axis z ----
  k_det<<<cdiv(BT, 256), 256, 0, stream>>>(pz, tri, dmat, 2);
  k_zero<<<cdiv(BN, 256), 256, 0, stream>>>(Cc, BN);
  k_gemm_wmma<<<gemm_grid, 256, 0, stream>>>(dmat, vf, Cc);
  k_solve<<<BB, 256, 0, stream>>>(Cc, x, aarr, defz, 2);
  k_update_add_z<<<cdiv(BN, 256), 256, 0, stream>>>(idx1, defz, pz);

  // ---- axis y ----
  k_det<<<cdiv(BT, 256), 256, 0, stream>>>(pz, tri, dmat, 1);
  k_zero<<<cdiv(BN, 256), 256, 0, stream>>>(Cc, BN);
  k_gemm_wmma<<<gemm_grid, 256, 0, stream>>>(dmat, vf, Cc);
  k_solve<<<BB, 256, 0, stream>>>(Cc, x, aarr, defy, 1);
  k_update_set_zy<<<cdiv(BN, 256), 256, 0, stream>>>(idx1, defy, pz);

  // ---- axis x ----
  k_det<<<cdiv(BT, 256), 256, 0, stream>>>(pz, tri, dmat, 0);
  k_zero<<<cdiv(BN, 256), 256, 0, stream>>>(Cc, BN);
  k_gemm_wmma<<<gemm_grid, 256, 0, stream>>>(dmat, vf, Cc);
  k_solve<<<BB, 256, 0, stream>>>(Cc, x, aarr, defx, 0);

  k_output<<<cdiv(BN, 256), 256, 0, stream>>>(x, defx, defy, defz, out);
  (void)in_sizes; (void)n_in; (void)out_size; (void)ws_size; (void)ggrid;
}